// TaylorEBM_50525995270536
// MI455X (gfx1250) — compile-verified
//
#include <hip/hip_runtime.h>

typedef __attribute__((ext_vector_type(2))) float v2f;
typedef __attribute__((ext_vector_type(8))) float v8f;

#define STEPS 50
#define LR 0.1f
#define HPAD 68   // 64 features + 4 pad (bank-conflict-free for stride-68 rows)

// ---------------------------------------------------------------------------
// Kernel 1: 1-D nearest neighbor (excluding self, first-min tie-break)
// ---------------------------------------------------------------------------
__global__ __launch_bounds__(256) void nn_kernel(const float* __restrict__ c,
                                                 float* __restrict__ c0,
                                                 int B) {
  __shared__ float tile[256];
  const int i = blockIdx.x * 256 + threadIdx.x;
  const float ci = c[i];
  float best = 3.0e38f;
  int bj = 0;
  for (int base = 0; base < B; base += 256) {
    __syncthreads();
    tile[threadIdx.x] = c[base + threadIdx.x];
    __syncthreads();
#pragma unroll 8
    for (int t = 0; t < 256; ++t) {
      const int j = base + t;
      float d = fabsf(ci - tile[t]);
      if (j == i) d += 1.0e9f;
      if (d < best) { best = d; bj = j; }
    }
  }
  c0[i] = c[bj];
}

// ---------------------------------------------------------------------------
// WMMA helpers (f32 16x16x4).
// A 16x4 layout : lane L<16 -> {A[L,0],A[L,1]}, lane L>=16 -> {A[L-16,2],A[L-16,3]}
// B 4x16 layout : lane L<16 -> {B[0,L],B[1,L]},  lane L>=16 -> {B[2,L-16],B[3,L-16]}
// C/D 16x16     : lane L, vgpr v -> C[v + 8*(L>=16), L%16]
// ---------------------------------------------------------------------------

// Z[fout, s] = relu( W @ H + bias ), H sample-major [16][HPAD], result same layout.
__device__ __forceinline__ void fwd_layer(const float* sWm, const float* sBias,
                                          const float* srcH, float* dstH,
                                          int lane) {
  const int sl = lane & 15;
  const int hi = lane >> 4;
#pragma unroll
  for (int m = 0; m < 4; ++m) {
    v8f acc;
#pragma unroll
    for (int v = 0; v < 8; ++v) acc[v] = sBias[16 * m + 8 * hi + v];
    const int arow = 16 * m + sl;
#pragma unroll
    for (int k = 0; k < 16; ++k) {
      const int kc = 4 * k + 2 * hi;
      v2f a, b;
      a.x = sWm[arow * 64 + kc];        // A = W[16m..+15, 4k..+3]
      a.y = sWm[arow * 64 + kc + 1];
      b.x = srcH[sl * HPAD + kc];       // B = H[4k..+3, samples]
      b.y = srcH[sl * HPAD + kc + 1];
      acc = __builtin_amdgcn_wmma_f32_16x16x4_f32(false, a, false, b,
                                                  (short)0, acc, false, false);
    }
#pragma unroll
    for (int v = 0; v < 8; ++v)
      dstH[sl * HPAD + 16 * m + 8 * hi + v] = fmaxf(acc[v], 0.0f);
  }
}

// Gprev[s, fin] = mask(H) * sum_fout D[s,fout] * W[fout,fin]   (uses W untransposed)
__device__ __forceinline__ void bwd_layer(const float* sWm, const float* srcD,
                                          const float* maskH, float* dstD,
                                          int lane) {
  const int sl = lane & 15;
  const int hi = lane >> 4;
#pragma unroll
  for (int n = 0; n < 4; ++n) {              // input-feature block
    v8f acc;
#pragma unroll
    for (int v = 0; v < 8; ++v) acc[v] = 0.0f;
#pragma unroll
    for (int k = 0; k < 16; ++k) {           // output-feature (K) block
      const int kc = 4 * k + 2 * hi;
      v2f a, b;
      a.x = srcD[sl * HPAD + kc];            // A = D^T[samples, fout]
      a.y = srcD[sl * HPAD + kc + 1];
      b.x = sWm[(kc + 0) * 64 + 16 * n + sl];// B = W[4k..+3, 16n..+15]
      b.y = sWm[(kc + 1) * 64 + 16 * n + sl];
      acc = __builtin_amdgcn_wmma_f32_16x16x4_f32(false, a, false, b,
                                                  (short)0, acc, false, false);
    }
#pragma unroll
    for (int v = 0; v < 8; ++v) {
      const int samp = v + 8 * hi;
      const int fin  = 16 * n + sl;
      const float g  = (maskH[samp * HPAD + fin] > 0.0f) ? acc[v] : 0.0f;
      dstD[samp * HPAD + fin] = g;
    }
  }
}

// ---------------------------------------------------------------------------
// Kernel 2: 50-step inner GD per sample, 16 samples per wave, one wave/block.
// ---------------------------------------------------------------------------
__global__ __launch_bounds__(32) void solve_kernel(
    const float* __restrict__ x, const float* __restrict__ c0g,
    const float* __restrict__ W1, const float* __restrict__ b1,
    const float* __restrict__ W2, const float* __restrict__ b2,
    const float* __restrict__ W3, const float* __restrict__ b3,
    const float* __restrict__ W4, float* __restrict__ out) {
  __shared__ float sW2[64 * 64];
  __shared__ float sW3[64 * 64];
  __shared__ float sW1[64 * 3];
  __shared__ float sW4[64];
  __shared__ float sB1[64], sB2[64], sB3[64];
  __shared__ float sH1[16 * HPAD], sH2[16 * HPAD], sH3[16 * HPAD], sX[16 * HPAD];
  __shared__ float sXin[16], sC0[16], sY[16];

  const int lane = threadIdx.x;
  const int s0 = blockIdx.x * 16;

  for (int idx = lane; idx < 64 * 64; idx += 32) {
    sW2[idx] = W2[idx];
    sW3[idx] = W3[idx];
  }
  for (int idx = lane; idx < 64 * 3; idx += 32) sW1[idx] = W1[idx];
  for (int idx = lane; idx < 64; idx += 32) {
    sW4[idx] = W4[idx];
    sB1[idx] = b1[idx];
    sB2[idx] = b2[idx];
    sB3[idx] = b3[idx];
  }
  if (lane < 16) {
    sXin[lane] = x[s0 + lane];
    sC0[lane]  = c0g[s0 + lane];
    sY[lane]   = 0.0f;   // Y_MEAN
  }
  __syncthreads();

  const int sl = lane & 15;
  const int hi = lane >> 4;
  const float xs = sXin[sl];
  const float cs = sC0[sl];

#pragma unroll 1
  for (int step = 0; step < STEPS; ++step) {
    // ---- layer 1: z1 = W1 @ [x,y,c0] + b1, relu ----
    const float ys = sY[sl];
#pragma unroll
    for (int f2 = 0; f2 < 64; f2 += 2) {
      const int fr = f2 + hi;
      float z = fmaf(sW1[fr * 3 + 0], xs,
                fmaf(sW1[fr * 3 + 1], ys,
                fmaf(sW1[fr * 3 + 2], cs, sB1[fr])));
      sH1[sl * HPAD + fr] = fmaxf(z, 0.0f);
    }
    __syncthreads();

    // ---- layers 2,3 forward (WMMA) ----
    fwd_layer(sW2, sB2, sH1, sH2, lane);
    __syncthreads();
    fwd_layer(sW3, sB3, sH2, sH3, lane);
    __syncthreads();

    // ---- seed: d3 = W4 * relu'(z3) ----
    for (int e = lane; e < 16 * 64; e += 32) {
      const int ss = e >> 6, f = e & 63;
      sX[ss * HPAD + f] = (sH3[ss * HPAD + f] > 0.0f) ? sW4[f] : 0.0f;
    }
    __syncthreads();

    // ---- backward (WMMA, sample-major, untransposed weights) ----
    bwd_layer(sW3, sX, sH2, sH3, lane);   // d2 -> reuse sH3
    __syncthreads();
    bwd_layer(sW2, sH3, sH1, sX, lane);   // d1 -> reuse sX
    __syncthreads();

    // ---- gy = W1[:,1]^T d1 ; y update ----
    float partial = 0.0f;
#pragma unroll
    for (int f = 0; f < 32; ++f) {
      const int fr = 32 * hi + f;
      partial = fmaf(sW1[fr * 3 + 1], sX[sl * HPAD + fr], partial);
    }
    partial += __shfl_down(partial, 16, 32);
    if (lane < 16) sY[sl] = ys - LR * partial;
    __syncthreads();
  }

  if (lane < 16) out[s0 + lane] = sY[lane];
}

// ---------------------------------------------------------------------------
extern "C" void kernel_launch(void* const* d_in, const int* in_sizes, int n_in,
                              void* d_out, int out_size, void* d_ws, size_t ws_size,
                              hipStream_t stream) {
  const float* x  = (const float*)d_in[0];
  const float* c  = (const float*)d_in[1];
  const float* W1 = (const float*)d_in[2];
  const float* b1 = (const float*)d_in[3];
  const float* W2 = (const float*)d_in[4];
  const float* b2 = (const float*)d_in[5];
  const float* W3 = (const float*)d_in[6];
  const float* b3 = (const float*)d_in[7];
  const float* W4 = (const float*)d_in[8];
  // d_in[9] = b4: constant offset, no effect on dE/dy — unused.

  const int B = in_sizes[0];         // 8192
  float* c0  = (float*)d_ws;         // 8192 floats of scratch
  float* out = (float*)d_out;

  nn_kernel<<<B / 256, 256, 0, stream>>>(c, c0, B);
  solve_kernel<<<B / 16, 32, 0, stream>>>(x, c0, W1, b1, W2, b2, W3, b3, W4, out);
}